// TreeLSTM_26912265077353
// MI455X (gfx1250) — compile-verified
//
#include <hip/hip_runtime.h>

typedef float v2f __attribute__((ext_vector_type(2)));
typedef float v4f __attribute__((ext_vector_type(4)));
typedef float v8f __attribute__((ext_vector_type(8)));

static constexpr int    kDepth = 13;
static constexpr int    kN     = (1 << kDepth) - 1;   // 8191 nodes per tree
static constexpr int    kB     = 32;
static constexpr int    kH     = 128;
static constexpr size_t kStack = (size_t)kB * kN * kH; // elements per output stack

// Padded LDS row strides (floats): keep 16B alignment for b128 stores and
// dodge the 64-bank conflict a 128/256-float stride would cause.
static constexpr int XS = 132;   // x rows       (K = 128)
static constexpr int HS = 260;   // h_cat rows   (K = 256)
static constexpr int CS = 260;   // child-c rows

__device__ __forceinline__ float sigf(float v) { return 1.0f / (1.0f + __expf(-v)); }

// One level of the tree. Block = 256 threads (8 wave32s) handles 16 rows of
// the (batch, node-in-level) space and all fused output columns:
//   cols [0,384)  : iou pre-activation  (x @ W_iou^T  [+ h_cat @ U_iou^T])
//   cols [384,640): f   pre-activation  (h_cat @ U_f_w^T)          (internal only)
template <bool LEAF>
__global__ __launch_bounds__(256) void tree_level(
    const float* __restrict__ x,     const float* __restrict__ W_iou,
    const float* __restrict__ U_iou, const float* __restrict__ b_iou,
    const float* __restrict__ U_f_w, const float* __restrict__ U_f_b,
    float* __restrict__ out,         // [2, B, N, H]; stack 1 doubles as c[] scratch
    int lo, int lsh)                 // level base node index, log2(n_l)
{
  constexpr int SOUT  = LEAF ? 384 : 640;  // fused output columns
  constexpr int NTILE = SOUT / 16;         // 24 or 40 column tiles
  constexpr int TPW   = NTILE / 8;         // tiles per wave: 3 or 5
  constexpr int OUTF  = 16 * SOUT;         // floats for the fused output slab
  constexpr int SMEMF = LEAF ? OUTF : (OUTF + 16 * CS);
  __shared__ float smem[SMEMF];

  float* ldsX = smem;              // 16 x XS   (phase 1/2)
  float* ldsH = smem + 16 * XS;    // 16 x HS   (phase 1/2, internal)
  float* ldsC = smem + OUTF;       // 16 x CS   (persists through phase 3, internal)
  float* ldsO = smem;              // 16 x SOUT (phase 3, overlays ldsX/ldsH)
  float* c_ws = out + kStack;      // c[] lives in output stack 1 during the sweep

  const int tid   = threadIdx.x;
  const int mbase = blockIdx.x * 16;
  const int jmask = (1 << lsh) - 1;

  // ---------------- Phase 1: stage A-operands into LDS ----------------
  for (int e = tid; e < 16 * 32; e += 256) {           // x rows: 16 x 32 float4
    int r = e >> 5, q = e & 31;
    int m = mbase + r, b = m >> lsh, node = lo + (m & jmask);
    const v4f* src = (const v4f*)(x + ((size_t)b * kN + node) * kH);
    *(v4f*)(ldsX + r * XS + q * 4) = src[q];
  }
  if constexpr (!LEAF) {
    for (int e = tid; e < 16 * 64; e += 256) {         // h_cat: 16 x 64 float4
      int r = e >> 6, q = e & 63;
      int m = mbase + r, b = m >> lsh, node = lo + (m & jmask);
      int child = 2 * node + 1 + (q >> 5);             // left then right child h
      const v4f* src = (const v4f*)(out + ((size_t)b * kN + child) * kH);
      *(v4f*)(ldsH + r * HS + q * 4) = src[q & 31];
    }
    for (int e = tid; e < 16 * 64; e += 256) {         // child c: 16 x 64 float4
      int r = e >> 6, q = e & 63;
      int m = mbase + r, b = m >> lsh, node = lo + (m & jmask);
      int child = 2 * node + 1 + (q >> 5);
      const v4f* src = (const v4f*)(c_ws + ((size_t)b * kN + child) * kH);
      *(v4f*)(ldsC + r * CS + q * 4) = src[q & 31];
    }
  }
  __syncthreads();

  // ---------------- Phase 2: WMMA GEMMs (fp32, 16x16x4) ----------------
  // A frag (16x4 f32): row = lane&15, VGPR pair holds K = k0+koff, k0+koff+1
  // B frag (4x16 f32): col = lane&15, same K sub-selection
  // D frag: VGPR r, lane L -> row r + 8*(L>>4), col L&15
  const int wave = tid >> 5, lane = tid & 31;
  const int arow = lane & 15;
  const int bcol = lane & 15;
  const int koff = (lane >> 4) * 2;

  v8f acc[TPW];
  #pragma unroll
  for (int u = 0; u < TPW; ++u) {
    const int T    = wave + 8 * u;     // wave-uniform: EXEC stays all-ones
    const int col0 = T * 16;
    v8f c = {};
    if (LEAF || T < 24) {
      // iou tile: x-part, K = 128
      const float* wp = W_iou + (size_t)(col0 + bcol) * 128;
      #pragma unroll 4
      for (int k0 = 0; k0 < 128; k0 += 4) {
        v2f a = *(const v2f*)(ldsX + arow * XS + k0 + koff);
        v2f b = *(const v2f*)(wp + k0 + koff);
        c = __builtin_amdgcn_wmma_f32_16x16x4_f32(false, a, false, b,
                                                  (short)0, c, false, false);
      }
      if constexpr (!LEAF) {           // + h_cat @ U_iou^T, K = 256
        const float* up = U_iou + (size_t)(col0 + bcol) * 256;
        #pragma unroll 4
        for (int k0 = 0; k0 < 256; k0 += 4) {
          v2f a = *(const v2f*)(ldsH + arow * HS + k0 + koff);
          v2f b = *(const v2f*)(up + k0 + koff);
          c = __builtin_amdgcn_wmma_f32_16x16x4_f32(false, a, false, b,
                                                    (short)0, c, false, false);
        }
      }
    } else {                           // f tile: h_cat @ U_f_w^T, K = 256
      const float* up = U_f_w + (size_t)(col0 - 384 + bcol) * 256;
      #pragma unroll 4
      for (int k0 = 0; k0 < 256; k0 += 4) {
        v2f a = *(const v2f*)(ldsH + arow * HS + k0 + koff);
        v2f b = *(const v2f*)(up + k0 + koff);
        c = __builtin_amdgcn_wmma_f32_16x16x4_f32(false, a, false, b,
                                                  (short)0, c, false, false);
      }
    }
    acc[u] = c;
  }
  __syncthreads();                     // all waves done reading ldsX/ldsH

  #pragma unroll
  for (int u = 0; u < TPW; ++u) {      // spill D frags into fused LDS slab
    const int col0 = (wave + 8 * u) * 16;
    #pragma unroll
    for (int r = 0; r < 8; ++r) {
      int rowi = r + 8 * (lane >> 4);
      ldsO[rowi * SOUT + col0 + bcol] = acc[u][r];
    }
  }
  __syncthreads();

  // ---------------- Phase 3: gate math + writeback ----------------
  for (int e = tid; e < 16 * kH; e += 256) {
    int r = e >> 7, hh = e & 127;
    int m = mbase + r, b = m >> lsh, node = lo + (m & jmask);
    const float* row = ldsO + r * SOUT;
    float iv = row[hh]        + b_iou[hh];
    float ov = row[128 + hh]  + b_iou[128 + hh];
    float uv = row[256 + hh]  + b_iou[256 + hh];
    float c_red = 0.0f;
    if constexpr (!LEAF) {
      float fl = sigf(row[384 + hh] + U_f_b[hh]);
      float fr = sigf(row[512 + hh] + U_f_b[128 + hh]);
      c_red = fl * ldsC[r * CS + hh] + fr * ldsC[r * CS + 128 + hh];
    }
    float cv = sigf(iv) * tanhf(uv) + c_red;
    float hv = sigf(ov) * tanhf(cv);
    size_t oi = ((size_t)b * kN + node) * kH + hh;
    c_ws[oi] = cv;           // consumed by the parent level
    out[oi]  = hv;           // stack 0
  }
}

// Final pass: stack 1 (which held c[] during the sweep) becomes a copy of
// the h[] in stack 0, matching the reference's jnp.stack([h] * 2).
__global__ __launch_bounds__(256) void dup_stack(float* __restrict__ out) {
  size_t i = (size_t)blockIdx.x * 256 + threadIdx.x;   // in float4 units
  const v4f* src = (const v4f*)out;
  v4f*       dst = (v4f*)(out + kStack);
  dst[i] = src[i];
}

extern "C" void kernel_launch(void* const* d_in, const int* /*in_sizes*/, int /*n_in*/,
                              void* d_out, int /*out_size*/, void* /*d_ws*/, size_t /*ws_size*/,
                              hipStream_t stream) {
  const float* x     = (const float*)d_in[0];
  const float* W_iou = (const float*)d_in[1];
  const float* U_iou = (const float*)d_in[2];
  const float* b_iou = (const float*)d_in[3];
  const float* U_f_w = (const float*)d_in[4];
  const float* U_f_b = (const float*)d_in[5];
  float* out = (float*)d_out;

  // Leaf level (l = 12): c_red = 0, no U terms.
  {
    int l = kDepth - 1, n_l = 1 << l, lo = n_l - 1;
    tree_level<true><<<dim3((kB * n_l) / 16), dim3(256), 0, stream>>>(
        x, W_iou, U_iou, b_iou, U_f_w, U_f_b, out, lo, l);
  }
  // Internal levels, bottom-up (sequential dependency).
  for (int l = kDepth - 2; l >= 0; --l) {
    int n_l = 1 << l, lo = n_l - 1;
    tree_level<false><<<dim3((kB * n_l) / 16), dim3(256), 0, stream>>>(
        x, W_iou, U_iou, b_iou, U_f_w, U_f_b, out, lo, l);
  }
  // Duplicate h into stack 1.
  dup_stack<<<dim3((unsigned)(kStack / 4 / 256)), dim3(256), 0, stream>>>(out);
}